// GNNEncoder_52922587021464
// MI455X (gfx1250) — compile-verified
//
#include <hip/hip_runtime.h>
#include <hip/hip_bf16.h>

typedef __attribute__((ext_vector_type(2))) float v2f;
typedef __attribute__((ext_vector_type(8))) float v8f;

#define CO 32      // channels per head
#define HC 128     // heads * channels

__device__ __forceinline__ unsigned ford(float f) {
  unsigned u = __float_as_uint(f);
  return (u & 0x80000000u) ? ~u : (u | 0x80000000u);
}
__device__ __forceinline__ float funord(unsigned u) {
  return __uint_as_float((u & 0x80000000u) ? (u ^ 0x80000000u) : ~u);
}

// ---------------- generic fp32 WMMA GEMM: C[M,NOUT] = A[M,K] * W[K,NOUT] ----
// NOUT in {32,128}. 256 threads = 8 waves. A tile staged in LDS, W via cache.
template <int NOUT>
__global__ void k_gemm(const float* __restrict__ A, const float* __restrict__ W,
                       float* __restrict__ Cm, int M, int K) {
  constexpr int NCT  = NOUT / 16;  // column tiles
  constexpr int RT   = 8 / NCT;    // row tiles per block
  constexpr int ROWS = RT * 16;
  extern __shared__ float lds_a[];  // ROWS * K floats
  const int tid = threadIdx.x;
  const int blockRow = blockIdx.x * ROWS;
  const int total = ROWS * K;
  for (int idx = tid; idx < total; idx += 256) {
    int r = idx / K, c = idx - r * K;
    int gr = blockRow + r;
    lds_a[idx] = (gr < M) ? A[(long long)gr * K + c] : 0.0f;
  }
  __syncthreads();
  const int wave = tid >> 5;
  const int lane = tid & 31;
  const int half = lane >> 4;     // 0: k = kk, kk+1; 1: k = kk+2, kk+3 (per ISA A/B layout)
  const int ml = lane & 15;
  const int rowTile = wave / NCT;
  const int colTile = wave - rowTile * NCT;
  const int r0 = rowTile * 16;
  const int gcol = colTile * 16 + ml;
  v8f acc = {};
  for (int kk = 0; kk < K; kk += 4) {
    const int ka0 = kk + (half ? 2 : 0);
    const int ka1 = kk + (half ? 3 : 1);
    v2f a, b;
    a.x = lds_a[(r0 + ml) * K + ka0];
    a.y = lds_a[(r0 + ml) * K + ka1];
    b.x = W[ka0 * NOUT + gcol];
    b.y = W[ka1 * NOUT + gcol];
    acc = __builtin_amdgcn_wmma_f32_16x16x4_f32(false, a, false, b, (short)0,
                                                acc, false, false);
  }
#pragma unroll
  for (int r = 0; r < 8; ++r) {
    int gr = blockRow + r0 + r + (half ? 8 : 0);
    if (gr < M) Cm[(long long)gr * NOUT + gcol] = acc[r];
  }
}

// ---------------- small helpers --------------------------------------------
__global__ void k_fill0(float* p, long long n) {
  long long i = (long long)blockIdx.x * blockDim.x + threadIdx.x;
  long long stride = (long long)gridDim.x * blockDim.x;
  for (; i < n; i += stride) p[i] = 0.0f;
}

// g[k*4+h] = sum_c We[k, h*32+c] * a[64+c]   (edge-attention vector, fused)
__global__ void k_gvec(const float* __restrict__ We, const float* __restrict__ a,
                       float* __restrict__ g) {
  int j = threadIdx.x;  // 256 threads
  int k = j >> 2, h = j & 3;
  float s = 0.0f;
  for (int c = 0; c < CO; ++c) s += We[k * HC + h * CO + c] * a[64 + c];
  g[j] = s;
}

// per (node, head): di = a[0:32].xp_h, dj = a[32:64].xp_h
__global__ void k_ndots(const float* __restrict__ xp, const float* __restrict__ a,
                        float* __restrict__ dn, long long N) {
  long long t = (long long)blockIdx.x * blockDim.x + threadIdx.x;
  if (t >= N * 4) return;
  long long n = t >> 2;
  int h = (int)(t & 3);
  const float* row = xp + n * HC + h * CO;
  float di = 0.0f, dj = 0.0f;
  for (int c = 0; c < CO; ++c) {
    float v = row[c];
    di += a[c] * v;
    dj += a[CO + c] * v;
  }
  dn[t * 2] = di;
  dn[t * 2 + 1] = dj;
}

// pass1: raw leaky_relu logits + segment max (ordered-uint atomicMax on dst)
__global__ void k_alpha1(const float* __restrict__ ea, const long long* __restrict__ src,
                         const long long* __restrict__ dst, const float* __restrict__ dn,
                         const float* __restrict__ g, float* __restrict__ alpha,
                         unsigned* __restrict__ m, long long E, long long N) {
  __shared__ float sg[256];
  sg[threadIdx.x] = g[threadIdx.x];
  __syncthreads();
  long long t = (long long)blockIdx.x * blockDim.x + threadIdx.x;
  if (t >= E + N) return;
  float de0 = 0, de1 = 0, de2 = 0, de3 = 0;
  long long s, d;
  if (t < E) {
    s = src[t];
    d = dst[t];
    const float* row = ea + t * 64;
    for (int k = 0; k < 64; ++k) {
      float v = row[k];
      de0 += v * sg[k * 4 + 0];
      de1 += v * sg[k * 4 + 1];
      de2 += v * sg[k * 4 + 2];
      de3 += v * sg[k * 4 + 3];
    }
  } else {
    s = d = t - E;  // self loop, edge features are zero
  }
  float de[4] = {de0, de1, de2, de3};
#pragma unroll
  for (int h = 0; h < 4; ++h) {
    float al = dn[d * 8 + h * 2] + dn[s * 8 + h * 2 + 1] + de[h];
    al = (al > 0.0f) ? al : 0.01f * al;  // leaky_relu
    alpha[t * 4 + h] = al;
    atomicMax(&m[d * 4 + h], ford(al));
  }
}

// pass2: exp(alpha - max) in place + segment sum
__global__ void k_alpha2(const long long* __restrict__ dst, float* __restrict__ alpha,
                         const unsigned* __restrict__ m, float* __restrict__ denom,
                         long long E, long long N) {
  long long t = (long long)blockIdx.x * blockDim.x + threadIdx.x;
  if (t >= E + N) return;
  long long d = (t < E) ? dst[t] : (t - E);
#pragma unroll
  for (int h = 0; h < 4; ++h) {
    float ex = __expf(alpha[t * 4 + h] - funord(m[d * 4 + h]));
    alpha[t * 4 + h] = ex;
    atomicAdd(&denom[d * 4 + h], ex);
  }
}

// self-loop messages: wave per node, lane = channel
__global__ void k_msg_self(const float* __restrict__ u, const float* __restrict__ alpha,
                           const float* __restrict__ denom, float* __restrict__ nodeEmb,
                           long long N, long long E) {
  long long gt = (long long)blockIdx.x * blockDim.x + threadIdx.x;
  long long i = gt >> 5;
  int c = (int)(gt & 31);
  if (i >= N) return;
  float acc = 0.0f;
#pragma unroll
  for (int h = 0; h < 4; ++h) {
    float w = alpha[(E + i) * 4 + h] / (denom[i * 4 + h] + 1e-16f) * 0.25f;
    acc += w * u[i * HC + h * CO + c];
  }
  atomicAdd(&nodeEmb[i * CO + c], acc);
}

// edge messages for a chunk: wave per edge, lane = channel
__global__ void k_msg_edges(const float* __restrict__ u, const float* __restrict__ vch,
                            const float* __restrict__ alpha, const float* __restrict__ denom,
                            const long long* __restrict__ src, const long long* __restrict__ dst,
                            float* __restrict__ nodeEmb, long long e0, long long M) {
  long long gt = (long long)blockIdx.x * blockDim.x + threadIdx.x;
  long long idx = gt >> 5;
  int c = (int)(gt & 31);
  if (idx >= M) return;
  long long e = e0 + idx;
  long long s = src[e], d = dst[e];
  float acc = 0.0f;
#pragma unroll
  for (int h = 0; h < 4; ++h) {
    float w = alpha[e * 4 + h] / (denom[d * 4 + h] + 1e-16f) * 0.25f;
    acc += w * (u[s * HC + h * CO + c] + vch[idx * HC + h * CO + c]);
  }
  atomicAdd(&nodeEmb[d * CO + c], acc);
}

__global__ void k_elu(const float* __restrict__ in, float* __restrict__ out, long long n) {
  long long i = (long long)blockIdx.x * blockDim.x + threadIdx.x;
  long long stride = (long long)gridDim.x * blockDim.x;
  for (; i < n; i += stride) {
    float v = in[i];
    out[i] = (v > 0.0f) ? v : (__expf(v) - 1.0f);
  }
}

// Wstack[h*32+c, c'] = Weu[64+c, c'] / 4   (for Wr = We @ Wstack)
__global__ void k_wstack(const float* __restrict__ Weu, float* __restrict__ ws) {
  int idx = blockIdx.x * blockDim.x + threadIdx.x;
  if (idx >= 128 * 32) return;
  int rc = idx >> 5;        // h*32+c
  int cp = idx & 31;
  int c = rc & 31;
  ws[idx] = 0.25f * Weu[(64 + c) * 32 + cp];
}

// final edge output: relu(p[src] + q[dst] + r)
__global__ void k_newe(const float* __restrict__ p, const float* __restrict__ q,
                       const float* __restrict__ rch, const long long* __restrict__ src,
                       const long long* __restrict__ dst, float* __restrict__ out,
                       long long e0, long long M) {
  long long gt = (long long)blockIdx.x * blockDim.x + threadIdx.x;
  long long idx = gt >> 5;
  int c = (int)(gt & 31);
  if (idx >= M) return;
  long long e = e0 + idx;
  long long s = src[e], d = dst[e];
  float v = p[s * CO + c] + q[d * CO + c] + rch[idx * CO + c];
  out[e * CO + c] = (v > 0.0f) ? v : 0.0f;
}

// ---------------- host side -------------------------------------------------
static inline void gemm128(const float* A, const float* W, float* C, int M, int K,
                           hipStream_t s) {
  k_gemm<128><<<(M + 15) / 16, 256, 16 * K * 4, s>>>(A, W, C, M, K);
}
static inline void gemm32(const float* A, const float* W, float* C, int M, int K,
                          hipStream_t s) {
  k_gemm<32><<<(M + 63) / 64, 256, 64 * K * 4, s>>>(A, W, C, M, K);
}

static void run_layer(const float* xin, int Kin, const float* Wn, const float* We,
                      const float* a, const float* Wen, const float* edge_attr,
                      const long long* src, const long long* dst, long long N, long long E,
                      float* xp, float* u, float* dn, unsigned* mbuf, float* denom,
                      float* nodeEmb, float* alpha, float* g, float* Wv, float* vch,
                      long long CH, hipStream_t s) {
  // zero m(4N) + denom(4N) + nodeEmb(32N), contiguous in the workspace
  k_fill0<<<1024, 256, 0, s>>>((float*)mbuf, N * 40);
  gemm128(xin, Wn, xp, (int)N, Kin, s);                 // xp = x @ Wn      [N,128]
  gemm32(We, Wen + CO * CO, Wv, 256, CO, s);            // Wv = We  @ Wen_bot (view [256,32])
  k_gvec<<<1, 256, 0, s>>>(We, a, g);                   // fused edge attention vector
  gemm32(xp, Wen, u, (int)(N * 4), CO, s);              // u = xp @ Wen_top (view [4N,32])
  k_ndots<<<(unsigned)((N * 4 + 255) / 256), 256, 0, s>>>(xp, a, dn, N);
  long long T = E + N;
  k_alpha1<<<(unsigned)((T + 255) / 256), 256, 0, s>>>(edge_attr, src, dst, dn, g, alpha,
                                                       mbuf, E, N);
  k_alpha2<<<(unsigned)((T + 255) / 256), 256, 0, s>>>(dst, alpha, mbuf, denom, E, N);
  k_msg_self<<<(unsigned)((N * 32 + 255) / 256), 256, 0, s>>>(u, alpha, denom, nodeEmb, N, E);
  for (long long e0 = 0; e0 < E; e0 += CH) {
    long long M = (E - e0 < CH) ? (E - e0) : CH;
    gemm128(edge_attr + e0 * 64, Wv, vch, (int)M, 64, s);  // v-chunk = ea @ Wv
    k_msg_edges<<<(unsigned)((M * 32 + 255) / 256), 256, 0, s>>>(u, vch, alpha, denom, src,
                                                                 dst, nodeEmb, e0, M);
  }
}

extern "C" void kernel_launch(void* const* d_in, const int* in_sizes, int n_in,
                              void* d_out, int out_size, void* d_ws, size_t ws_size,
                              hipStream_t stream) {
  const float* x         = (const float*)d_in[0];
  const float* edge_attr = (const float*)d_in[1];
  const long long* eidx  = (const long long*)d_in[2];  // int64 per reference
  const float* Wn1  = (const float*)d_in[3];
  const float* We1  = (const float*)d_in[4];
  const float* a1   = (const float*)d_in[5];
  const float* Wen1 = (const float*)d_in[6];
  // d_in[7] = Weu1: layer-1 edge output is dead code in the reference; skipped.
  const float* Wn2  = (const float*)d_in[8];
  const float* We2  = (const float*)d_in[9];
  const float* a2   = (const float*)d_in[10];
  const float* Wen2 = (const float*)d_in[11];
  const float* Weu2 = (const float*)d_in[12];

  const long long N = in_sizes[0] / 128;
  const long long E = in_sizes[1] / 64;
  const long long* src = eidx;
  const long long* dst = eidx + E;
  const long long CH = 131072;

  float* ws = (float*)d_ws;
  long long o = 0;
  float* xp = ws + o;        o += N * 128;
  float* u  = ws + o;        o += N * 128;
  float* dn = ws + o;        o += N * 8;
  unsigned* mbuf = (unsigned*)(ws + o); o += N * 4;   // m, denom, nodeEmb contiguous
  float* denom   = ws + o;   o += N * 4;
  float* nodeEmb = ws + o;   o += N * 32;
  float* x1   = ws + o;      o += N * 32;
  float* pbuf = ws + o;      o += N * 32;
  float* qbuf = ws + o;      o += N * 32;
  float* alpha = ws + o;     o += (E + N) * 4;
  float* g   = ws + o;       o += 256;
  float* Wv  = ws + o;       o += 256 * 32;
  float* Wst = ws + o;       o += 128 * 32;
  float* Wr  = ws + o;       o += 64 * 32;
  float* vch = ws + o;       o += CH * 128;
  float* rch = ws + o;       o += CH * 32;

  float* outNode = (float*)d_out;              // elu(x2) [N,32]
  float* outEdge = (float*)d_out + N * 32;     // relu(e2) [E,32]

  // ---- layer 1 (edge-update skipped: dead in reference) ----
  run_layer(x, 128, Wn1, We1, a1, Wen1, edge_attr, src, dst, N, E, xp, u, dn, mbuf,
            denom, nodeEmb, alpha, g, Wv, vch, CH, stream);
  k_elu<<<2048, 256, 0, stream>>>(nodeEmb, x1, N * 32);

  // ---- layer 2 (uses ORIGINAL edge_attr, per reference) ----
  run_layer(x1, 32, Wn2, We2, a2, Wen2, edge_attr, src, dst, N, E, xp, u, dn, mbuf,
            denom, nodeEmb, alpha, g, Wv, vch, CH, stream);

  // node output
  k_elu<<<2048, 256, 0, stream>>>(nodeEmb, outNode, N * 32);

  // edge update for layer 2
  k_wstack<<<16, 256, 0, stream>>>(Weu2, Wst);
  gemm32(We2, Wst, Wr, 64, 128, stream);                 // Wr = We2 @ stack(Weu2[2C:3C])/H
  gemm32(nodeEmb, Weu2, pbuf, (int)N, 32, stream);       // p = node_emb @ Weu[0:C]
  gemm32(nodeEmb, Weu2 + 32 * 32, qbuf, (int)N, 32, stream);  // q = node_emb @ Weu[C:2C]
  for (long long e0 = 0; e0 < E; e0 += CH) {
    long long M = (E - e0 < CH) ? (E - e0) : CH;
    gemm32(edge_attr + e0 * 64, Wr, rch, (int)M, 64, stream);  // r-chunk = ea @ Wr
    k_newe<<<(unsigned)((M * 32 + 255) / 256), 256, 0, stream>>>(pbuf, qbuf, rch, src, dst,
                                                                 outEdge, e0, M);
  }
}